// OTNetwork_65721589563621
// MI455X (gfx1250) — compile-verified
//
#include <hip/hip_runtime.h>
#include <hip/hip_bf16.h>
#include <math.h>

typedef __attribute__((ext_vector_type(2))) float v2f;
typedef __attribute__((ext_vector_type(8))) float v8f;
typedef __attribute__((ext_vector_type(4))) int   v4i;

#define GLOBAL_AS __attribute__((address_space(1)))
#define LDS_AS    __attribute__((address_space(3)))

#define TILE_M 64
#define TILE_N 128
#define TILE_K 32
#define A_STRIDE 36  // words; 144B pitch: 16B aligned for b128 async, conflict-free b64 frags
#define B_STRIDE 34  // words; Bs[col][k] transposed: 8B aligned b64 frags, conflict-free

#if defined(__AMDGCN__) && __has_builtin(__builtin_amdgcn_global_load_async_to_lds_b128) && \
    __has_builtin(__builtin_amdgcn_global_load_async_to_lds_b32)
#define HAVE_ASYNC 1
#else
#define HAVE_ASYNC 0
#endif

#if HAVE_ASYNC
#define WAIT_ASYNC(n) asm volatile("s_wait_asynccnt " #n ::: "memory")
#else
#define WAIT_ASYNC(n)
#endif

// Issue one K-slab stage:
//   A rows [blockM,+64) x k[kb,+32) -> asBuf[row][k]   (2 x b128 per thread)
//   B rows k[kb,+32) x cols [blockN,+128) -> bsBuf[col][k] TRANSPOSED scatter
//     (16 x b32 per thread; per-lane LDS dest addressing of the async engine)
// Every thread issues exactly 18 async copies per stage.
__device__ __forceinline__ void stage_tiles(
    const float* __restrict__ A, const float* __restrict__ B,
    float* asBuf, float* bsBuf, int tid, int blockM, int blockN,
    int kb, int K, int N) {
#if HAVE_ASYNC
#pragma unroll
  for (int t = 0; t < 2; ++t) {
    const int i = tid + t * 256;          // 0..511
    const int r = i >> 3, q = i & 7;
    const float* g = A + (size_t)(blockM + r) * K + kb + q * 4;
    float* l = asBuf + r * A_STRIDE + q * 4;
    __builtin_amdgcn_global_load_async_to_lds_b128(
        (GLOBAL_AS v4i*)g, (LDS_AS v4i*)l, 0, 0);
  }
#pragma unroll
  for (int t = 0; t < 16; ++t) {
    const int i = tid + t * 256;          // 0..4095
    const int k = i >> 7, n = i & 127;    // coalesced in n; scattered in LDS
    const float* g = B + (size_t)(kb + k) * N + blockN + n;
    float* l = bsBuf + n * B_STRIDE + k;
    __builtin_amdgcn_global_load_async_to_lds_b32(
        (GLOBAL_AS int*)g, (LDS_AS int*)l, 0, 0);
  }
#else
  // Synchronous fallback, identical layout.
#pragma unroll
  for (int t = 0; t < 2; ++t) {
    const int i = tid + t * 256;
    const int r = i >> 3, q = i & 7;
    const float4 v = *(const float4*)(A + (size_t)(blockM + r) * K + kb + q * 4);
    *(float4*)(asBuf + r * A_STRIDE + q * 4) = v;
  }
#pragma unroll
  for (int t = 0; t < 16; ++t) {
    const int i = tid + t * 256;
    const int k = i >> 7, n = i & 127;
    bsBuf[n * B_STRIDE + k] = B[(size_t)(kb + k) * N + blockN + n];
  }
#endif
}

// ---------------------------------------------------------------------------
// GEMM: C = [relu](A @ B + bias), A:(M,K) row-major, B:(K,N) row-major.
// 256 threads = 8 waves; block computes 64x128 of C; wave = 32x32 (2x2 WMMA
// 16x16 tiles). Double-buffered LDS; async global->LDS staging issued one slab
// ahead (ASYNCcnt in-order retirement -> s_wait_asynccnt 18 = current slab
// landed). Transposed B in LDS makes every WMMA operand a contiguous
// ds_load_2addr_b64 pair (no packing movs).
// WMMA f32 16x16x4 per-lane fragments: A = {A[row][k0], A[row][k0+1]},
// B = {B[k0][col], B[k0+1][col]}, row/col = lane&15, k0 = 2*(lane>>4) (+4*kk).
// ---------------------------------------------------------------------------
template <bool RELU>
__global__ __launch_bounds__(256) void gemm_bias_kernel(
    const float* __restrict__ A, const float* __restrict__ B,
    const float* __restrict__ bias, float* __restrict__ C,
    int M, int N, int K) {
  __shared__ float As[2][TILE_M * A_STRIDE];   // As[buf][row][k]
  __shared__ float Bs[2][TILE_N * B_STRIDE];   // Bs[buf][col][k] (transposed)

  const int tid   = threadIdx.x;
  const int lane  = tid & 31;
  const int wave  = tid >> 5;
  const int waveM = wave & 1;    // 0..1 -> 32-row strip
  const int waveN = wave >> 1;   // 0..3 -> 32-col strip
  const int blockM = blockIdx.y * TILE_M;
  const int blockN = blockIdx.x * TILE_N;

  const int l16 = lane & 15;
  const int lhi = lane >> 4;     // 0 or 1
  const int arow0 = waveM * 32 + l16;
  const int arow1 = arow0 + 16;
  const int bcol0 = waveN * 32 + l16;
  const int bcol1 = bcol0 + 16;

  v8f c00 = {};
  v8f c01 = {};
  v8f c10 = {};
  v8f c11 = {};

  // Prologue: stage first slab into buffer 0.
  stage_tiles(A, B, As[0], Bs[0], tid, blockM, blockN, 0, K, N);

  int buf = 0;
  for (int kb = 0; kb < K; kb += TILE_K) {
    const bool hasNext = (kb + TILE_K) < K;
    if (hasNext) {
      // Previous compute on buf^1 was barriered last iteration -> safe to fill.
      stage_tiles(A, B, As[buf ^ 1], Bs[buf ^ 1], tid, blockM, blockN,
                  kb + TILE_K, K, N);
      WAIT_ASYNC(18);  // retire current slab's 18 copies (in-order); next 18 in flight
    } else {
      WAIT_ASYNC(0);
    }
    __syncthreads();   // current slab visible to all waves

    const float* as = &As[buf][0];
    const float* bs = &Bs[buf][0];
#pragma unroll
    for (int kk = 0; kk < 8; ++kk) {
      const int k0 = kk * 4 + lhi * 2;
      const v2f a0 = *(const v2f*)&as[arow0 * A_STRIDE + k0];
      const v2f a1 = *(const v2f*)&as[arow1 * A_STRIDE + k0];
      const v2f b0 = *(const v2f*)&bs[bcol0 * B_STRIDE + k0];
      const v2f b1 = *(const v2f*)&bs[bcol1 * B_STRIDE + k0];
      c00 = __builtin_amdgcn_wmma_f32_16x16x4_f32(false, a0, false, b0,
                                                  (short)0, c00, false, false);
      c01 = __builtin_amdgcn_wmma_f32_16x16x4_f32(false, a0, false, b1,
                                                  (short)0, c01, false, false);
      c10 = __builtin_amdgcn_wmma_f32_16x16x4_f32(false, a1, false, b0,
                                                  (short)0, c10, false, false);
      c11 = __builtin_amdgcn_wmma_f32_16x16x4_f32(false, a1, false, b1,
                                                  (short)0, c11, false, false);
    }
    __syncthreads();   // all waves done reading buf before it is re-staged
    buf ^= 1;
  }

  // ---- Epilogue: C/D layout -> lane<16: VGPR j = M j ; lane>=16: M j+8.
  const int col0 = blockN + waveN * 32 + l16;
  const int col1 = col0 + 16;
  const float bias0 = bias[col0];
  const float bias1 = bias[col1];
#pragma unroll
  for (int j = 0; j < 8; ++j) {
    const int row0 = blockM + waveM * 32 + j + 8 * lhi;
    const int row1 = row0 + 16;
    float v00 = c00[j] + bias0;
    float v01 = c01[j] + bias1;
    float v10 = c10[j] + bias0;
    float v11 = c11[j] + bias1;
    if (RELU) {
      v00 = fmaxf(v00, 0.0f);
      v01 = fmaxf(v01, 0.0f);
      v10 = fmaxf(v10, 0.0f);
      v11 = fmaxf(v11, 0.0f);
    }
    C[(size_t)row0 * N + col0] = v00;
    C[(size_t)row0 * N + col1] = v01;
    C[(size_t)row1 * N + col0] = v10;
    C[(size_t)row1 * N + col1] = v11;
  }
}

// ---------------------------------------------------------------------------
// Sinkhorn: one block per batch item, 64x64 matrix resident in LDS (stride 65
// to avoid bank conflicts). In-place: reads cost, writes plan. Matches
// reference: P=exp(-g*M); P/=sum(P); loop<=1000: row-normalize, col sums,
// err=max|beta-1/64|; break BEFORE col-normalizing when err<=eps.
// ---------------------------------------------------------------------------
#define SK_GAMMA 1.0f
#define SK_EPS   1e-6f
#define SK_ITERS 1000

__global__ __launch_bounds__(256) void sinkhorn_kernel(float* __restrict__ P) {
  __shared__ float Ps[64 * 65];
  __shared__ float part[256];
  __shared__ float alpha[64];
  __shared__ float beta[64];

  const int tid = threadIdx.x;
  float* base = P + (size_t)blockIdx.x * 4096;

  float lsum = 0.0f;
  for (int i = tid; i < 4096; i += 256) {
    const int r = i >> 6, c = i & 63;
    const float v = __expf(-SK_GAMMA * base[i]);
    Ps[r * 65 + c] = v;
    lsum += v;
  }
  part[tid] = lsum;
  __syncthreads();
  for (int s = 128; s > 0; s >>= 1) {
    if (tid < s) part[tid] += part[tid + s];
    __syncthreads();
  }
  const float inv = 1.0f / part[0];
  __syncthreads();
  for (int i = tid; i < 4096; i += 256) {
    const int r = i >> 6, c = i & 63;
    Ps[r * 65 + c] *= inv;
  }
  __syncthreads();

  const float marg = 1.0f / 64.0f;
  const int idx = tid >> 2;
  const int q   = tid & 3;

  for (int it = 0; it < SK_ITERS; ++it) {
    float s = 0.0f;
#pragma unroll
    for (int c = 0; c < 16; ++c) s += Ps[idx * 65 + q * 16 + c];
    part[tid] = s;
    __syncthreads();
    if (q == 0) alpha[idx] = part[tid] + part[tid + 1] + part[tid + 2] + part[tid + 3];
    __syncthreads();
    for (int i = tid; i < 4096; i += 256) {
      const int r = i >> 6, c = i & 63;
      Ps[r * 65 + c] *= marg / alpha[r];
    }
    __syncthreads();
    float t = 0.0f;
#pragma unroll
    for (int r = 0; r < 16; ++r) t += Ps[(q * 16 + r) * 65 + idx];
    part[tid] = t;
    __syncthreads();
    if (q == 0) beta[idx] = part[tid] + part[tid + 1] + part[tid + 2] + part[tid + 3];
    __syncthreads();
    if (tid < 64) part[tid] = fabsf(beta[tid] - marg);
    __syncthreads();
    for (int s2 = 32; s2 > 0; s2 >>= 1) {
      if (tid < s2) part[tid] = fmaxf(part[tid], part[tid + s2]);
      __syncthreads();
    }
    const float err = part[0];
    __syncthreads();
    if (err <= SK_EPS) break;
    for (int i = tid; i < 4096; i += 256) {
      const int r = i >> 6, c = i & 63;
      Ps[r * 65 + c] *= marg / beta[c];
    }
    __syncthreads();
  }

  for (int i = tid; i < 4096; i += 256) {
    const int r = i >> 6, c = i & 63;
    base[i] = Ps[r * 65 + c];
  }
}

// ---------------------------------------------------------------------------
extern "C" void kernel_launch(void* const* d_in, const int* in_sizes, int n_in,
                              void* d_out, int out_size, void* d_ws, size_t ws_size,
                              hipStream_t stream) {
  const float* z  = (const float*)d_in[0];  // (2048, 256)
  const float* W1 = (const float*)d_in[1];  // (256, 4096)
  const float* b1 = (const float*)d_in[2];  // (4096)
  const float* W2 = (const float*)d_in[3];  // (4096, 4096)
  const float* b2 = (const float*)d_in[4];  // (4096)
  const float* W3 = (const float*)d_in[5];  // (4096, 4096)
  const float* b3 = (const float*)d_in[6];  // (4096)

  const int Mb = 2048, H = 4096, DZ = 256;

  float* X1 = (float*)d_ws;                       // (2048, 4096)
  float* X2 = X1 + (size_t)Mb * H;                // (2048, 4096)
  float* X3 = (float*)d_out;                      // (2048, 4096) == (2048,64,64)

  dim3 blk(256);
  dim3 grid(H / TILE_N, Mb / TILE_M);             // (32, 32)

  gemm_bias_kernel<true ><<<grid, blk, 0, stream>>>(z,  W1, b1, X1, Mb, H, DZ);
  gemm_bias_kernel<true ><<<grid, blk, 0, stream>>>(X1, W2, b2, X2, Mb, H, H);
  gemm_bias_kernel<false><<<grid, blk, 0, stream>>>(X2, W3, b3, X3, Mb, H, H);
  sinkhorn_kernel<<<Mb, blk, 0, stream>>>(X3);
}